// FNOBlock_31464930411022
// MI455X (gfx1250) — compile-verified
//
#include <hip/hip_runtime.h>

// FNO block for MI455X (gfx1250, wave32, WMMA).
// out = v + irfft( scatter( einsum('mio,bmi->bmo', R, rfft(v)[:, :512]) ) )
// restructured as three batched fp32 WMMA GEMMs (truncated DFT as matmul),
// each a double-buffered LDS-staged block GEMM: global b128 loads for chunk
// i+1 are issued before chunk i's compute (latency hidden under 64 WMMAs),
// then ds_store'd into the alternate LDS buffer.

typedef float v2f __attribute__((ext_vector_type(2)));
typedef float v4f __attribute__((ext_vector_type(4)));
typedef float v8f __attribute__((ext_vector_type(8)));

#define PI_D 3.14159265358979323846

// ---------------------------------------------------------------------------
// DFT table generation (double-precision on device; exact integer phase mod N)
// Wf: (1024 x 4096)  row k<512 : cos(2*pi*k*n/4096)
//                    row 512+k : -sin(2*pi*k*n/4096)        (Re/Im of rfft)
// Wi: (4096 x 1024)  col k<512 : ((k==0)?1:2)/4096 * cos(2*pi*k*n/4096)
//                    col 512+k : -(2/4096) * sin(2*pi*k*n/4096)
// ---------------------------------------------------------------------------
__global__ void gen_dft_tables(float* __restrict__ Wf, float* __restrict__ Wi) {
  long long idx = (long long)blockIdx.x * blockDim.x + threadIdx.x;
  const long long total = 1024LL * 4096LL;
  if (idx >= total) return;

  {  // Wf entry: k = idx/4096, n = idx%4096
    int k = (int)(idx >> 12), n = (int)(idx & 4095);
    int kk = k & 511;
    int t = (kk * n) & 4095;                 // exact phase mod N
    double th = (2.0 * PI_D / 4096.0) * (double)t;
    Wf[idx] = (k < 512) ? (float)cos(th) : (float)(-sin(th));
  }
  {  // Wi entry: n = idx/1024, k = idx%1024
    int n = (int)(idx >> 10), k = (int)(idx & 1023);
    int kk = k & 511;
    int t = (kk * n) & 4095;
    double th = (2.0 * PI_D / 4096.0) * (double)t;
    double val;
    if (k < 512) {
      double scale = (kk == 0) ? (1.0 / 4096.0) : (2.0 / 4096.0);
      val = scale * cos(th);
    } else {
      val = -(2.0 / 4096.0) * sin(th);
    }
    Wi[idx] = (float)val;
  }
}

// ---------------------------------------------------------------------------
// Double-buffered LDS-staged batched GEMM, fp32 WMMA (V_WMMA_F32_16X16X4_F32).
//   D_batch = A_batch @ B_batch (+ Src_batch)      all fp32 row-major
//
// Block = 128 threads = 4 waves (WM x WN layout, WM*WN==4).
// Block tile: BM=32*WM rows x BN=64*WN cols. Each wave owns a 32x64 sub-tile
// (2x4 accumulators of 16x16; 8 WMMAs per K=4 step, A regs reused across N).
// K is processed in chunks of 32. Pipeline per chunk:
//   issue global b128 loads for chunk i+1 (regs) -> compute chunk i from LDS
//   -> barrier -> ds_store regs into alternate buffer -> barrier.
//
// ISA VGPR layouts (CDNA5 CDNA5 ISA):
//   A 16x4  (MxK): lane = (k>=2)*16 + m , vgpr = k&1   -> 2 floats per lane
//   B 4x16  (KxN): lane = (k>=2)*16 + n , vgpr = k&1
//   C/D 16x16:     vgpr j holds M=j (lanes 0-15) / M=j+8 (lanes 16-31), N=lane&15
// ---------------------------------------------------------------------------
template <int WM, int WN>
__global__ __launch_bounds__(128) void wmma_gemm_lds(
    const float* __restrict__ A, const float* __restrict__ B,
    float* __restrict__ D, const float* __restrict__ Src,
    int K,
    long long lda, long long ldb, long long ldd,
    long long strideA, long long strideB, long long strideD, long long strideSrc,
    int nBlocks, int tilesPerBatch) {
  constexpr int BM = 32 * WM;        // block rows
  constexpr int BN = 64 * WN;        // block cols
  constexpr int KC = 32;             // K chunk
  constexpr int AST = KC + 4;        // padded LDS stride for A (16B aligned)
  constexpr int BST = BN + 4;        // padded LDS stride for B (16B aligned)
  constexpr int NA = WM * 2;         // A float4s per thread per chunk
  constexpr int NB = WN * 4;         // B float4s per thread per chunk

  __shared__ float lAs[2][BM * AST];
  __shared__ float lBs[2][KC * BST];

  const int tid = threadIdx.x;
  const int lane = tid & 31;
  const int wave = tid >> 5;
  const int wm = wave % WM;          // wave position along M
  const int wn = wave / WM;          // wave position along N

  const int batch = blockIdx.x / tilesPerBatch;
  const int rem = blockIdx.x % tilesPerBatch;
  const int mb = rem / nBlocks;
  const int nb = rem % nBlocks;

  const float* Ab = A + (long long)batch * strideA;
  const float* Bb = B + (long long)batch * strideB;
  float* Db = D + (long long)batch * strideD;

  const int m0b = mb * BM;           // block row origin
  const int n0b = nb * BN;           // block col origin
  const int l15 = lane & 15;
  const int lhi = lane >> 4;         // 0/1 : K-halves {0,1} / {2,3}

  // Per-thread staging coordinates (fixed across chunks).
  int aRow[NA], aCol[NA];
#pragma unroll
  for (int i = 0; i < NA; ++i) {     // BM*8 float4 over 128 threads
    int q = tid + 128 * i;
    aRow[i] = q >> 3;
    aCol[i] = (q & 7) * 4;
  }
  int bRow[NB], bCol[NB];
#pragma unroll
  for (int i = 0; i < NB; ++i) {     // 8*BN float4 over 128 threads
    int q = tid + 128 * i;
    bRow[i] = q / (BN / 4);
    bCol[i] = (q % (BN / 4)) * 4;
  }

  v8f acc[2][4];
#pragma unroll
  for (int mi = 0; mi < 2; ++mi)
#pragma unroll
    for (int ni = 0; ni < 4; ++ni)
      acc[mi][ni] = (v8f)(0.0f);

  // ---- prologue: stage chunk 0 into buffer 0 ----
#pragma unroll
  for (int i = 0; i < NA; ++i) {
    v4f val = *(const v4f*)(Ab + (long long)(m0b + aRow[i]) * lda + aCol[i]);
    *(v4f*)&lAs[0][aRow[i] * AST + aCol[i]] = val;
  }
#pragma unroll
  for (int i = 0; i < NB; ++i) {
    v4f val = *(const v4f*)(Bb + (long long)bRow[i] * ldb + n0b + bCol[i]);
    *(v4f*)&lBs[0][bRow[i] * BST + bCol[i]] = val;
  }
  __syncthreads();

  const int nChunks = K / KC;
  const int aOff0 = (wm * 32 + l15) * AST;
  const int aOff1 = (wm * 32 + 16 + l15) * AST;

  for (int ci = 0; ci < nChunks; ++ci) {
    const int p = ci & 1;
    const bool hasNext = (ci + 1 < nChunks);   // block-uniform

    // ---- issue global loads for chunk ci+1 (latency hidden by compute) ----
    v4f pa[NA], pb[NB];
    if (hasNext) {
      const int kc1 = (ci + 1) * KC;
#pragma unroll
      for (int i = 0; i < NA; ++i)
        pa[i] = *(const v4f*)(Ab + (long long)(m0b + aRow[i]) * lda + kc1 + aCol[i]);
#pragma unroll
      for (int i = 0; i < NB; ++i)
        pb[i] = *(const v4f*)(Bb + (long long)(kc1 + bRow[i]) * ldb + n0b + bCol[i]);
    }

    // ---- compute chunk ci from LDS buffer p: 8 K-steps x 8 WMMAs ----
#pragma unroll
    for (int k = 0; k < KC; k += 4) {
      v2f a0 = *(const v2f*)&lAs[p][aOff0 + k + 2 * lhi];
      v2f a1 = *(const v2f*)&lAs[p][aOff1 + k + 2 * lhi];
#pragma unroll
      for (int ni = 0; ni < 4; ++ni) {
        int col = wn * 64 + ni * 16 + l15;
        v2f b;
        b.x = lBs[p][(k + 2 * lhi) * BST + col];
        b.y = lBs[p][(k + 2 * lhi + 1) * BST + col];
        acc[0][ni] = __builtin_amdgcn_wmma_f32_16x16x4_f32(
            false, a0, false, b, (short)0, acc[0][ni], false, false);
        acc[1][ni] = __builtin_amdgcn_wmma_f32_16x16x4_f32(
            false, a1, false, b, (short)0, acc[1][ni], false, false);
      }
    }

    // ---- commit prefetched chunk into the alternate buffer ----
    if (hasNext) {
      __syncthreads();   // all waves done reading buffer p^1 (prev iteration)
#pragma unroll
      for (int i = 0; i < NA; ++i)
        *(v4f*)&lAs[p ^ 1][aRow[i] * AST + aCol[i]] = pa[i];
#pragma unroll
      for (int i = 0; i < NB; ++i)
        *(v4f*)&lBs[p ^ 1][bRow[i] * BST + bCol[i]] = pb[i];
      __syncthreads();   // stores visible before next compute
    }
  }

  // ---- epilogue: optional residual add (Src shares ldd), store ----
  const float* Sb = Src ? (Src + (long long)batch * strideSrc) : nullptr;
#pragma unroll
  for (int mi = 0; mi < 2; ++mi) {
#pragma unroll
    for (int ni = 0; ni < 4; ++ni) {
#pragma unroll
      for (int j = 0; j < 8; ++j) {
        int row = m0b + wm * 32 + mi * 16 + j + 8 * lhi;
        int col = n0b + wn * 64 + ni * 16 + l15;
        long long off = (long long)row * ldd + col;
        float val = acc[mi][ni][j];
        if (Sb) val += Sb[off];
        Db[off] = val;
      }
    }
  }
}

// ---------------------------------------------------------------------------
// Host launcher. Workspace (floats):
//   Wf : 1024*4096            (16 MB)
//   Wi : 4096*1024            (16 MB)
//   F  : 32*1024*256          (32 MB)  per-b (1024x256): rows 0..511 Re, 512..1023 Im
//   G  : 32*1024*256          (32 MB)
// ---------------------------------------------------------------------------
extern "C" void kernel_launch(void* const* d_in, const int* in_sizes, int n_in,
                              void* d_out, int out_size, void* d_ws, size_t ws_size,
                              hipStream_t stream) {
  const float* v = (const float*)d_in[0];   // (32, 4096, 256)
  const float* R = (const float*)d_in[1];   // (512, 256, 256)
  float* out = (float*)d_out;               // (32, 4096, 256)

  float* ws = (float*)d_ws;
  float* Wf = ws;
  float* Wi = Wf + 1024LL * 4096;
  float* F  = Wi + 4096LL * 1024;
  float* G  = F + 32LL * 1024 * 256;

  // 1) DFT tables
  {
    long long total = 1024LL * 4096;
    int threads = 256;
    int blocks = (int)((total + threads - 1) / threads);
    gen_dft_tables<<<blocks, threads, 0, stream>>>(Wf, Wi);
  }

  // 2) Forward truncated rfft:  F_b (1024x256) = Wf (1024x4096) @ v_b (4096x256)
  //    block tile 128x64: mBlocks=8, nBlocks=4, batch=32 -> 1024 blocks
  wmma_gemm_lds<4, 1><<<1024, 128, 0, stream>>>(
      Wf, v, F, nullptr,
      /*K=*/4096,
      /*lda=*/4096, /*ldb=*/256, /*ldd=*/256,
      /*strideA=*/0, /*strideB=*/4096LL * 256, /*strideD=*/1024LL * 256,
      /*strideSrc=*/0,
      /*nBlocks=*/4, /*tilesPerBatch=*/8 * 4);

  // 3) Mode mixing: for each m (batch=512):
  //      A_m (64x256) rows r=(2b+im) at F + r*(512*256) + m*256  (lda=131072)
  //      B_m = R + m*65536 (256x256), D_m rows at G + r*131072 + m*256
  //    block tile 64x128: mBlocks=1, nBlocks=2 -> 1024 blocks
  wmma_gemm_lds<2, 2><<<1024, 128, 0, stream>>>(
      F, R, G, nullptr,
      /*K=*/256,
      /*lda=*/512LL * 256, /*ldb=*/256, /*ldd=*/512LL * 256,
      /*strideA=*/256, /*strideB=*/256LL * 256, /*strideD=*/256,
      /*strideSrc=*/0,
      /*nBlocks=*/2, /*tilesPerBatch=*/1 * 2);

  // 4) Inverse irfft + residual: out_b (4096x256) = v_b + Wi (4096x1024) @ G_b (1024x256)
  //    block tile 128x64: mBlocks=32, nBlocks=4, batch=32 -> 4096 blocks
  wmma_gemm_lds<4, 1><<<4096, 128, 0, stream>>>(
      Wi, G, out, v,
      /*K=*/1024,
      /*lda=*/1024, /*ldb=*/256, /*ldd=*/256,
      /*strideA=*/0, /*strideB=*/1024LL * 256, /*strideD=*/4096LL * 256,
      /*strideSrc=*/4096LL * 256,
      /*nBlocks=*/4, /*tilesPerBatch=*/32 * 4);

  (void)in_sizes; (void)n_in; (void)out_size; (void)ws_size;
}